// MPSTPCP_32066225832325
// MI455X (gfx1250) — compile-verified
//
#include <hip/hip_runtime.h>
#include <hip/hip_bf16.h>
#include <math.h>

// MPS-TPCP chain, CDNA5 (gfx1250, wave32) WMMA formulation.
//
// Per layer l, per element b:  red = G_l : (vec(rho) (x) vec(x x^T)),
// G_l[4][16] precomputed from kraus & W.  Mapped to V_WMMA_F32_16X16X4_F32:
//   A (16x4 f32)  = G_l chunk (rows 0-3 = G rows, rows 8-11 = duplicate,
//                   others 0) -> D rows 0-3 / 8-11 give red in BOTH lane halves
//   B (4x16 f32)  = per-element rho[c] * (x x^T)[kk] (built with 8 v_mul)
//   4 chained WMMAs accumulate the K=16 contraction.
// Scale invariance of the final ratio lets us drop input normalization and
// fold the trace renorm into the B operand scale. The trace renorm is a pure
// scale factor that cancels in the final o0/(o0+o1), so a 1-ulp v_rcp_f32 is
// exact w.r.t. the final output; it removes ~10 dependent VALU ops/layer from
// the critical chain vs the IEEE divide sequence.

typedef float v2f __attribute__((ext_vector_type(2)));
typedef float v8f __attribute__((ext_vector_type(8)));

#define KOPS 4

// ---------------------------------------------------------------------------
// Prep: build WMMA-A-layout G tensors. GA layout: [l][c(4)][lane(32)][v(2)]
// A 16x4 f32 layout: lane<16: M=lane, VGPR0->K=0, VGPR1->K=1;
//                    lane>=16: M=lane-16, VGPR0->K=2, VGPR1->K=3.
// ---------------------------------------------------------------------------
__global__ __launch_bounds__(256) void mps_prep_GA(
    const float* __restrict__ kraus, const float* __restrict__ W,
    float* __restrict__ GA) {
  const int l = blockIdx.x;
  __shared__ float G[4][16];  // [p=2j+jj][t=4q+r]
  const float* A = kraus + (size_t)l * KOPS * 16;
  const float* w = W + (size_t)l * 4;

  const int t0 = threadIdx.x;
  if (t0 < 64) {
    const int p = t0 >> 4, t = t0 & 15;
    const int j = p >> 1, jj = p & 1;
    const int q = t >> 2, r = t & 3;
    const int bh = q >> 1, ch = q & 1, bl = r >> 1, cl = r & 1;
    float s = 0.f;
    for (int k = 0; k < KOPS; ++k) {
      const float* Ak = A + k * 16;
      for (int i = 0; i < 2; ++i)
        for (int m = 0; m < 2; ++m)
          s += w[i * 2 + m] * Ak[(2 * m + j) * 4 + (2 * bh + bl)] *
               Ak[(2 * i + jj) * 4 + (2 * ch + cl)];
    }
    G[p][t] = s;
  }
  __syncthreads();

  // 256 threads -> 256 GA floats for this layer. t1 = c*64 + lane*2 + v.
  const int t1 = threadIdx.x;
  const int c = t1 >> 6;
  const int lane = (t1 >> 1) & 31;
  const int v = t1 & 1;
  const int M = lane & 15;
  const int Kk = v + ((lane >= 16) ? 2 : 0);
  float val = 0.f;
  int pr = -1;
  if (M < 4) pr = M;                      // real rows
  else if (M >= 8 && M < 12) pr = M - 8;  // duplicated rows -> lane-half mirror
  if (pr >= 0) val = G[pr][4 * c + Kk];
  GA[(size_t)l * 256 + t1] = val;
}

// ---------------------------------------------------------------------------
// Main chain: one wave handles 16 batch elements (WMMA columns), both lane
// halves carry the same element (lane & 15).
// ---------------------------------------------------------------------------
__global__ __launch_bounds__(256) void mps_main(
    const float* __restrict__ X, const float* __restrict__ GA,
    const float* __restrict__ r, float* __restrict__ out,
    int B, int N, int L, int nWaves) {
  const int wave = (blockIdx.x * blockDim.x + threadIdx.x) >> 5;
  const int lane = threadIdx.x & 31;
  if (wave >= nWaves) return;  // wave-uniform exit; survivors have full EXEC

  const int e = wave * 16 + (lane & 15);
  const int ec = (e < B) ? e : (B - 1);                 // clamp for loads only
  const float* xrow = X + (size_t)ec * N * 2;
  const bool hi = (lane >= 16);
  const float* gl = GA + (size_t)lane * 2;              // + l*256 + c*64

  // rho_vec (row-major 2x2), unnormalized outer(x0); trace-renorm folds in.
  float2 x0 = *(const float2*)(xrow);
  float4 rho = make_float4(x0.x * x0.x, x0.x * x0.y, x0.y * x0.x, x0.y * x0.y);

  // software pipeline: prefetch layer-0 operands
  float2 xn = *(const float2*)(xrow + 2);
  v2f a0n = *(const v2f*)(gl + 0);
  v2f a1n = *(const v2f*)(gl + 64);
  v2f a2n = *(const v2f*)(gl + 128);
  v2f a3n = *(const v2f*)(gl + 192);

  for (int l = 0; l < L; ++l) {
    const float2 x = xn;
    const v2f a0 = a0n, a1 = a1n, a2 = a2n, a3 = a3n;
    // prefetch next layer (redundant reload on last iter keeps it uniform)
    const int ln = (l + 1 < L) ? (l + 1) : l;
    xn = *(const float2*)(xrow + 2 * (ln + 1));
    const float* gn = gl + (size_t)ln * 256;
    a0n = *(const v2f*)(gn + 0);
    a1n = *(const v2f*)(gn + 64);
    a2n = *(const v2f*)(gn + 128);
    a3n = *(const v2f*)(gn + 192);

    // B operand: b_c = rho[c] * { xx^T[K_lo], xx^T[K_hi] } * 1/trace(rho).
    // Assumed B 4x16 layout mirrors A: VGPR0 = {K=0 | K=2}, VGPR1 = {K=1 | K=3}.
    const float tr = rho.x + rho.w;
    const float inv = __builtin_amdgcn_rcpf(fmaxf(tr, 1e-12f));  // v_rcp_f32
    const float xs = hi ? x.y : x.x;           // row selector of xx^T
    const float u0 = xs * x.x * inv;           // xx^T[K= 0 or 2] / tr
    const float u1 = xs * x.y * inv;           // xx^T[K= 1 or 3] / tr

    const v8f z = {0.f, 0.f, 0.f, 0.f, 0.f, 0.f, 0.f, 0.f};
    v2f b;
    b.x = rho.x * u0; b.y = rho.x * u1;
    v8f acc = __builtin_amdgcn_wmma_f32_16x16x4_f32(false, a0, false, b,
                                                    (short)0, z, false, false);
    b.x = rho.y * u0; b.y = rho.y * u1;
    acc = __builtin_amdgcn_wmma_f32_16x16x4_f32(false, a1, false, b,
                                                (short)0, acc, false, false);
    b.x = rho.z * u0; b.y = rho.z * u1;
    acc = __builtin_amdgcn_wmma_f32_16x16x4_f32(false, a2, false, b,
                                                (short)0, acc, false, false);
    b.x = rho.w * u0; b.y = rho.w * u1;
    acc = __builtin_amdgcn_wmma_f32_16x16x4_f32(false, a3, false, b,
                                                (short)0, acc, false, false);
    // D rows 0-3 (lanes<16) == rows 8-11 (lanes>=16): both halves updated.
    rho.x = acc[0]; rho.y = acc[1]; rho.z = acc[2]; rho.w = acc[3];
  }

  // Readout: m0 = (r e0)(r e0)^T, m1 = (r e1)(r e1)^T; all scalings cancel.
  if (lane < 16 && e < B) {
    const float c00 = r[0], c01 = r[1], c10 = r[2], c11 = r[3];
    const float off = rho.y + rho.z;
    const float o0 = c00 * c00 * rho.x + c00 * c10 * off + c10 * c10 * rho.w;
    const float o1 = c01 * c01 * rho.x + c01 * c11 * off + c11 * c11 * rho.w;
    out[e] = o0 / (o0 + o1);
  }
}

// ---------------------------------------------------------------------------
extern "C" void kernel_launch(void* const* d_in, const int* in_sizes, int n_in,
                              void* d_out, int out_size, void* d_ws,
                              size_t ws_size, hipStream_t stream) {
  const float* X     = (const float*)d_in[0];  // (B, N, 2)
  const float* kraus = (const float*)d_in[1];  // (L, K, 4, 4)
  const float* r     = (const float*)d_in[2];  // (2, 2)
  const float* W     = (const float*)d_in[3];  // (L, 2, 2)

  const int L = in_sizes[3] / 4;       // W = L*2*2
  const int N = L + 1;
  const int B = in_sizes[0] / (N * 2);

  float* GA = (float*)d_ws;            // L * 256 floats (~802 KB), L2-resident

  mps_prep_GA<<<L, 256, 0, stream>>>(kraus, W, GA);

  const int nWaves = (B + 15) / 16;
  const int threads = nWaves * 32;
  const int block = 256;
  const int grid = (threads + block - 1) / block;
  mps_main<<<grid, block, 0, stream>>>(X, GA, r, (float*)d_out, B, N, L,
                                       nWaves);
}